// BodyModel_11433202942515
// MI455X (gfx1250) — compile-verified
//
#include <hip/hip_runtime.h>

// ---------------- problem sizes ----------------
#define NB   512                 // batch
#define NJ   55                  // joints
#define NV   10475               // vertices
#define NS   16                  // betas
#define PF   486                 // (NJ-1)*9 pose feature
#define KX   504                 // PF + NS + kid + ones (extended GEMM K)
#define K1   480                 // phase-1 K (pure posedirs, 120 steps)
#define KE   24                  // Aext columns: K in [480,504)
#define M3   31425               // NV*3
#define M3P  31440               // padded to 16
#define MT3  1965                // M3P/16
#define NT3  32                  // NB/16
#define KSK  56                  // padded 55 for skinning K
#define NSK  8192                // NB*16 (cc padded 12->16)
#define VTIL 655                 // ceil(NV/16)
#define NVP  10480               // padded vertices for packed weights

// d_out layout: joints [0,84480) | vertices [84480,16174080) | glob [16174080,16427520)
#define OUT_VERT 84480
#define OUT_GLOB 16174080

__constant__ int c_par[NJ] = {
    -1,0,0,0,1,2,3,4,5,6,7,8,9,9,9,12,13,14,16,17,18,19,15,15,15,
    20,25,26,20,28,29,20,31,32,20,34,35,20,37,38,21,40,41,21,43,44,
    21,46,47,21,49,50,21,52,53 };

typedef __attribute__((ext_vector_type(2))) float v2f;
typedef __attribute__((ext_vector_type(8))) float v8f;

// pair-interleaved B layout: element (k, n) lives at [(k>>1)*2*NCOLS + n*2 + (k&1)]
__device__ __forceinline__ void bpair_write(float* base, int ncols, int k, int n, float v) {
    base[(size_t)(k >> 1) * (2 * ncols) + n * 2 + (k & 1)] = v;
}

// ---------------- kernel 1: Rodrigues + extended feature matrix F^T ----------------
// FTp: pair-interleaved (KX x NB). rows [0,486)=pose_feature, [486,502)=betas, 502=kid, 503=1.
__global__ void k_rodrigues(const float* __restrict__ rv,
                            const float* __restrict__ betas,
                            const float* __restrict__ kid,
                            float* __restrict__ rel,
                            float* __restrict__ FTp)
{
    int t = blockIdx.x * blockDim.x + threadIdx.x;
    if (t >= NB * NJ) return;
    int b = t / NJ, jj = t % NJ;

    float x = rv[t*3+0], y = rv[t*3+1], z = rv[t*3+2];
    float ang = sqrtf(x*x + y*y + z*z);
    float inv = 1.0f / fmaxf(ang, 1e-8f);
    float ax = x*inv, ay = y*inv, az = z*inv;
    float s = sinf(ang), c = cosf(ang), C = 1.0f - c;

    float R[9];
    R[0] = c + ax*ax*C;      R[1] = ax*ay*C - az*s;  R[2] = ax*az*C + ay*s;
    R[3] = ay*ax*C + az*s;   R[4] = c + ay*ay*C;     R[5] = ay*az*C - ax*s;
    R[6] = az*ax*C - ay*s;   R[7] = az*ay*C + ax*s;  R[8] = c + az*az*C;

    #pragma unroll
    for (int e = 0; e < 9; e++) rel[(b*NJ + jj)*9 + e] = R[e];

    if (jj > 0) {
        int k0 = (jj - 1) * 9;
        #pragma unroll
        for (int e = 0; e < 9; e++) bpair_write(FTp, NB, k0 + e, b, R[e]);
    } else {
        #pragma unroll
        for (int ss = 0; ss < NS; ss++) bpair_write(FTp, NB, PF + ss, b, betas[b*NS + ss]);
        bpair_write(FTp, NB, PF + NS,     b, kid[b]);
        bpair_write(FTp, NB, PF + NS + 1, b, 1.0f);
    }
}

// ---------------- kernel 2: kinematic chain (one thread per batch) ----------------
__global__ void k_kinematics(const float* __restrict__ rel,
                             const float* __restrict__ betas,
                             const float* __restrict__ kid,
                             const float* __restrict__ Jt,
                             const float* __restrict__ Jsd,
                             const float* __restrict__ kJsd,
                             const float* __restrict__ trn,
                             float* __restrict__ out,
                             float* __restrict__ transl)
{
    int b = blockIdx.x * blockDim.x + threadIdx.x;
    if (b >= NB) return;

    float bt[NS];
    #pragma unroll
    for (int s = 0; s < NS; s++) bt[s] = betas[b*NS + s];
    float kf = kid[b];

    float jl[NJ][3];
    for (int i = 0; i < NJ; i++)
        for (int c = 0; c < 3; c++) {
            float v = Jt[i*3 + c];
            const float* row = Jsd + (i*3 + c)*NS;
            #pragma unroll
            for (int s = 0; s < NS; s++) v += row[s]*bt[s];
            jl[i][c] = v + kJsd[i*3 + c]*kf;
        }

    float g[NJ][9];
    const float* rb = rel + b*NJ*9;
    #pragma unroll
    for (int e = 0; e < 9; e++) g[0][e] = rb[e];
    for (int i = 1; i < NJ; i++) {
        int p = c_par[i];
        const float* r = rb + i*9;
        for (int rr = 0; rr < 3; rr++)
            for (int cc = 0; cc < 3; cc++)
                g[i][rr*3+cc] = g[p][rr*3+0]*r[0*3+cc]
                              + g[p][rr*3+1]*r[1*3+cc]
                              + g[p][rr*3+2]*r[2*3+cc];
    }

    float pos[NJ][3];
    pos[0][0]=jl[0][0]; pos[0][1]=jl[0][1]; pos[0][2]=jl[0][2];
    for (int i = 1; i < NJ; i++) {
        int p = c_par[i];
        float bx = jl[i][0]-jl[p][0], by = jl[i][1]-jl[p][1], bz = jl[i][2]-jl[p][2];
        for (int rr = 0; rr < 3; rr++)
            pos[i][rr] = pos[p][rr] + g[p][rr*3+0]*bx + g[p][rr*3+1]*by + g[p][rr*3+2]*bz;
    }

    float tx = trn[b*3+0], ty = trn[b*3+1], tz = trn[b*3+2];
    for (int i = 0; i < NJ; i++) {
        for (int rr = 0; rr < 3; rr++) {
            float gj = g[i][rr*3+0]*jl[i][0] + g[i][rr*3+1]*jl[i][1] + g[i][rr*3+2]*jl[i][2];
            transl[(b*NJ + i)*3 + rr] = pos[i][rr] - gj;
        }
        out[(b*NJ + i)*3 + 0] = pos[i][0] + tx;
        out[(b*NJ + i)*3 + 1] = pos[i][1] + ty;
        out[(b*NJ + i)*3 + 2] = pos[i][2] + tz;
        #pragma unroll
        for (int e = 0; e < 9; e++) out[OUT_GLOB + (b*NJ + i)*9 + e] = g[i][e];
    }
}

// ---------------- kernel 3: pack skinning B-matrix T^T (pair-interleaved KSK x NSK) --
__global__ void k_packT(const float* __restrict__ out,
                        const float* __restrict__ transl,
                        float* __restrict__ TTp)
{
    int t = blockIdx.x * blockDim.x + threadIdx.x;
    if (t >= KSK * NB) return;
    int jj = t / NB, b = t % NB;
    float vals[16];
    #pragma unroll
    for (int i = 0; i < 16; i++) vals[i] = 0.0f;
    if (jj < NJ) {
        #pragma unroll
        for (int e = 0; e < 9; e++) vals[e] = out[OUT_GLOB + (b*NJ + jj)*9 + e];
        #pragma unroll
        for (int c = 0; c < 3; c++) vals[9 + c] = transl[(b*NJ + jj)*3 + c];
    }
    #pragma unroll
    for (int i = 0; i < 16; i++) bpair_write(TTp, NSK, jj, b*16 + i, vals[i]);
}

// ---------------- kernel 4a: pack A-extension (M3P x 24), K in [480,504) ----------
__global__ void k_packAext(const float* __restrict__ pd, const float* __restrict__ sd,
                           const float* __restrict__ vt, const float* __restrict__ ksd,
                           float* __restrict__ Aext)
{
    int t = blockIdx.x * blockDim.x + threadIdx.x;
    if (t >= M3P * KE) return;
    int m = t / KE, kk = t % KE;
    int k = K1 + kk;
    float v = 0.0f;
    if (m < M3) {
        v = (k < PF)      ? pd[(size_t)m*PF + k]
          : (k < PF + NS) ? sd[m*NS + (k - PF)]
          : (k == PF+NS)  ? ksd[m]
          :                 vt[m];
    }
    Aext[t] = v;
}

// ---------------- kernel 4b: pack weights zero-padded (NVP x KSK) ------------------
__global__ void k_packW(const float* __restrict__ wts, float* __restrict__ Wp)
{
    int t = blockIdx.x * blockDim.x + threadIdx.x;
    if (t >= NVP * KSK) return;
    int v = t / KSK, k = t % KSK;
    Wp[t] = (v < NV && k < NJ) ? wts[v*NJ + k] : 0.0f;
}

// ---------------- kernel 5: WMMA GEMM -> v_posed (M3 x NB) ----------------
__global__ void k_vposed(const float* __restrict__ pd, const float* __restrict__ Aext,
                         const float* __restrict__ FTp, float* __restrict__ vposed)
{
    int wave = blockIdx.x * (blockDim.x >> 5) + (threadIdx.x >> 5);
    int lane = threadIdx.x & 31;
    int mt = wave / NT3, nt = wave % NT3;
    int m0 = mt * 16;
    int hi = lane >> 4, lo = lane & 15;
    int mA = m0 + lo;                         // A row (lane%16)
    int mAc = (mA < M3) ? mA : (M3 - 1);      // clamp, no branch
    int nB = nt * 16 + lo;                    // B/C/D column

    v8f acc = {0.f,0.f,0.f,0.f,0.f,0.f,0.f,0.f};

    // phase 1: K in [0,480), dense unguarded posedirs, aligned float2 loads
    const float* ap = pd + (size_t)mAc * PF + hi*2;
    const float* bp = FTp + (size_t)hi * (2*NB) + nB*2;
    #pragma unroll 4
    for (int s = 0; s < K1/4; s++) {
        v2f a  = *(const v2f*)ap;
        v2f bb = *(const v2f*)bp;
        acc = __builtin_amdgcn_wmma_f32_16x16x4_f32(false, a, false, bb,
                                                    (short)0, acc, false, false);
        ap += 4;
        bp += 4*NB;   // two pair-rows per step
    }

    // phase 2: K in [480,504) from packed extension (padded rows, no guards)
    const float* ap2 = Aext + (size_t)mA * KE + hi*2;
    const float* bp2 = FTp + (size_t)(K1 >> 1) * (2*NB) + (size_t)hi * (2*NB) + nB*2;
    #pragma unroll
    for (int s = 0; s < KE/4; s++) {
        v2f a  = *(const v2f*)ap2;
        v2f bb = *(const v2f*)bp2;
        acc = __builtin_amdgcn_wmma_f32_16x16x4_f32(false, a, false, bb,
                                                    (short)0, acc, false, false);
        ap2 += 4;
        bp2 += 4*NB;
    }

    #pragma unroll
    for (int r = 0; r < 8; r++) {
        int m = m0 + r + hi*8;                // C/D: lanes16-31 hold M = r+8
        if (m < M3) vposed[(size_t)m*NB + nB] = acc[r];
    }
}

// ---------------- kernel 6: WMMA skinning GEMM + fused LBS epilogue ----------------
__global__ void k_skin(const float* __restrict__ Wp, const float* __restrict__ TTp,
                       const float* __restrict__ vposed, const float* __restrict__ trn,
                       float* __restrict__ out)
{
    __shared__ float lds[8][256];
    int wv = threadIdx.x >> 5;
    int lane = threadIdx.x & 31;
    int wave = blockIdx.x * 8 + wv;
    int mt = wave / NB, b = wave % NB;        // one 16-vertex tile x one batch per wave
    int m0 = mt * 16;
    int hi = lane >> 4, lo = lane & 15;
    int vA = m0 + lo;                         // < NVP, padded weights -> unguarded

    v8f acc = {0.f,0.f,0.f,0.f,0.f,0.f,0.f,0.f};
    const float* ap = Wp + (size_t)vA * KSK + hi*2;
    const float* bp = TTp + (size_t)hi * (2*NSK) + (b*16 + lo)*2;
    #pragma unroll
    for (int s = 0; s < KSK/4; s++) {
        v2f a  = *(const v2f*)ap;
        v2f bb = *(const v2f*)bp;
        acc = __builtin_amdgcn_wmma_f32_16x16x4_f32(false, a, false, bb,
                                                    (short)0, acc, false, false);
        ap += 4;
        bp += 4*NSK;
    }

    // stage tile (16 vertices x 16 cc) into LDS
    #pragma unroll
    for (int r = 0; r < 8; r++) {
        int mm = r + hi*8;
        lds[wv][mm*16 + lo] = acc[r];
    }
    __syncthreads();

    if (lane < 16) {
        int vtx = m0 + lane;
        if (vtx < NV) {
            const float* M = &lds[wv][lane*16];   // 9 rot + 3 transl
            float p0 = vposed[(size_t)(vtx*3 + 0)*NB + b];
            float p1 = vposed[(size_t)(vtx*3 + 1)*NB + b];
            float p2 = vposed[(size_t)(vtx*3 + 2)*NB + b];
            float tx = trn[b*3+0], ty = trn[b*3+1], tz = trn[b*3+2];
            float* o = out + OUT_VERT + ((size_t)b*NV + vtx)*3;
            o[0] = M[0]*p0 + M[1]*p1 + M[2]*p2 + M[9]  + tx;
            o[1] = M[3]*p0 + M[4]*p1 + M[5]*p2 + M[10] + ty;
            o[2] = M[6]*p0 + M[7]*p1 + M[8]*p2 + M[11] + tz;
        }
    }
}

// ---------------- launcher ----------------
extern "C" void kernel_launch(void* const* d_in, const int* in_sizes, int n_in,
                              void* d_out, int out_size, void* d_ws, size_t ws_size,
                              hipStream_t stream) {
    const float* rv    = (const float*)d_in[0];   // pose_rotvecs (B,J,3)
    const float* betas = (const float*)d_in[1];   // shape_betas (B,S)
    const float* trn   = (const float*)d_in[2];   // trans (B,3)
    const float* kid   = (const float*)d_in[3];   // kid_factor (B,)
    const float* vt    = (const float*)d_in[4];   // v_template (V,3)
    const float* sd    = (const float*)d_in[5];   // shapedirs (V,3,S)
    const float* pd    = (const float*)d_in[6];   // posedirs (V,3,486)
    const float* Jt    = (const float*)d_in[7];   // J_template (J,3)
    const float* Jsd   = (const float*)d_in[8];   // J_shapedirs (J,3,S)
    const float* ksd   = (const float*)d_in[9];   // kid_shapedir (V,3)
    const float* kJsd  = (const float*)d_in[10];  // kid_J_shapedir (J,3)
    const float* wts   = (const float*)d_in[11];  // weights (V,J)
    float* out = (float*)d_out;

    float* ws     = (float*)d_ws;
    float* rel    = ws;                   // NB*NJ*9  = 253440
    float* FTp    = rel    + 253440;      // KX*NB    = 258048
    float* transl = FTp    + 258048;      // NB*NJ*3  = 84480
    float* TTp    = transl + 84480;       // KSK*NSK  = 458752
    float* Wp     = TTp    + 458752;      // NVP*KSK  = 586880
    float* Aext   = Wp     + 586880;      // M3P*KE   = 754560
    float* vposed = Aext   + 754560;      // M3*NB    = 16089600  (~74 MB total)

    k_rodrigues <<<(NB*NJ + 255)/256, 256, 0, stream>>>(rv, betas, kid, rel, FTp);
    k_kinematics<<<2, 256, 0, stream>>>(rel, betas, kid, Jt, Jsd, kJsd, trn, out, transl);
    k_packT     <<<(KSK*NB + 255)/256, 256, 0, stream>>>(out, transl, TTp);
    k_packAext  <<<(M3P*KE + 255)/256, 256, 0, stream>>>(pd, sd, vt, ksd, Aext);
    k_packW     <<<(NVP*KSK + 255)/256, 256, 0, stream>>>(wts, Wp);
    k_vposed    <<<(MT3*NT3)/8, 256, 0, stream>>>(pd, Aext, FTp, vposed);
    k_skin      <<<(VTIL*NB)/8, 256, 0, stream>>>(Wp, TTp, vposed, trn, out);
}